// PolicyGraphConvNet_34514357190985
// MI455X (gfx1250) — compile-verified
//
#include <hip/hip_runtime.h>
#include <hip/hip_bf16.h>

// ---------------------------------------------------------------------------
// PolicyGraphConvNet: 4-layer GCN (jump concat) on MI455X / gfx1250.
//   per layer: agg = scatter_add(norm * h[row] -> col)
//              z   = [agg | h] @ W + b   (W: [2*d, dout], row-major)
//              h   = relu(z)  (last layer: no relu, dout=3)
// Dense updates use V_WMMA_F32_16X16X4_F32 (exact fp32, matrix pipe).
// Scatter uses hardware global_atomic_add_f32 (L2-resident working set).
// ---------------------------------------------------------------------------

#define N_NODES 50000
#define N_EDGES 800000
#define DIM     128
#define KDIM    (2 * DIM)   // 256
#define OUTD    3
#define LDA     260         // LDS row pitch (floats): 260*4 B = 65*16 B -> 16B aligned
                            // rows, bank-conflict-free strided reads.

typedef __attribute__((ext_vector_type(2))) float v2f;
typedef __attribute__((ext_vector_type(8))) float v8f;

// ---------------------------------------------------------------- degree ----
__global__ void deg_kernel(const int* __restrict__ col, float* __restrict__ deg) {
    int e = blockIdx.x * blockDim.x + threadIdx.x;
    if (e < N_EDGES) unsafeAtomicAdd(&deg[col[e]], 1.0f);
}

__global__ void dis_kernel(float* __restrict__ deg_dis) {
    int i = blockIdx.x * blockDim.x + threadIdx.x;
    if (i < N_NODES) {
        float d = deg_dis[i];
        deg_dis[i] = (d > 0.0f) ? rsqrtf(fmaxf(d, 1.0f)) : 0.0f;
    }
}

__global__ void norm_kernel(const int* __restrict__ row, const int* __restrict__ col,
                            const float* __restrict__ dis, float* __restrict__ norm) {
    int e = blockIdx.x * blockDim.x + threadIdx.x;
    if (e < N_EDGES) norm[e] = dis[row[e]] * dis[col[e]];
}

// ------------------------------------------------------- edge scatter-add ----
// One wave (32 lanes) per edge; each lane moves a float4 -> 128 features.
__global__ void scatter_kernel(const float* __restrict__ hin,
                               const float* __restrict__ norm,
                               const int* __restrict__ row,
                               const int* __restrict__ col,
                               float* __restrict__ agg) {
    int tid  = blockIdx.x * blockDim.x + threadIdx.x;
    int e    = tid >> 5;
    int lane = tid & 31;
    if (e >= N_EDGES) return;
    float nv = norm[e];
    int   r  = row[e];
    int   c  = col[e];
    const float4* src = reinterpret_cast<const float4*>(hin + (size_t)r * DIM);
    float4 v = src[lane];
    float* dst = agg + (size_t)c * DIM + lane * 4;
    unsafeAtomicAdd(dst + 0, nv * v.x);
    unsafeAtomicAdd(dst + 1, nv * v.y);
    unsafeAtomicAdd(dst + 2, nv * v.z);
    unsafeAtomicAdd(dst + 3, nv * v.w);
}

// --------------------------------------------------- WMMA f32 GEMM (d=128) ---
// Block: 256 threads (8 waves). Block tile: M=16 rows x N=128 cols.
// A tile (16 x 256, concat agg|h) staged in LDS; each wave owns a 16-col slice.
// V_WMMA_F32_16X16X4_F32 per-lane layout (ISA 7.12.2):
//   A: M = lane&15, K = kbase + 2*(lane>>4) + reg   (2 VGPRs)
//   B: N = lane&15, K = kbase + 2*(lane>>4) + reg   (2 VGPRs)
//   D: row = r + 8*(lane>>4), col = lane&15         (8 VGPRs)
__global__ void __launch_bounds__(256)
gemm_kernel(const float* __restrict__ agg, const float* __restrict__ hin,
            const float* __restrict__ W,   const float* __restrict__ bias,
            float* __restrict__ out, int relu) {
    __shared__ float Atile[16 * LDA];

    const int m0  = blockIdx.x * 16;
    const int t   = threadIdx.x;

    // Cooperative A-tile load: thread t loads 16 contiguous floats of
    // row (t>>4), segment (t&15). Cols [0,128) from agg, [128,256) from hin.
    {
        int r    = t >> 4;
        int c0   = (t & 15) * 16;
        int node = m0 + r;
        const float* src = (c0 < DIM) ? (agg + (size_t)node * DIM + c0)
                                      : (hin + (size_t)node * DIM + (c0 - DIM));
        #pragma unroll
        for (int i = 0; i < 16; i += 4) {
            float4 v = *reinterpret_cast<const float4*>(src + i);
            *reinterpret_cast<float4*>(&Atile[r * LDA + c0 + i]) = v;
        }
    }
    __syncthreads();

    const int wave = t >> 5;
    const int lane = t & 31;
    const int hi   = lane >> 4;
    const int nc   = wave * 16 + (lane & 15);
    const float* arow = &Atile[(lane & 15) * LDA];

    v8f acc = {0.f, 0.f, 0.f, 0.f, 0.f, 0.f, 0.f, 0.f};

    #pragma unroll 8
    for (int k = 0; k < KDIM; k += 4) {
        int kk = k + 2 * hi;
        v2f a, b;
        a.x = arow[kk];
        a.y = arow[kk + 1];
        b.x = W[(size_t)kk * DIM + nc];
        b.y = W[(size_t)(kk + 1) * DIM + nc];
        acc = __builtin_amdgcn_wmma_f32_16x16x4_f32(
                /*neg_a=*/false, a, /*neg_b=*/false, b,
                /*c_mod=*/(short)0, acc, /*reuse_a=*/false, /*reuse_b=*/false);
    }

    float bv = bias[nc];
    #pragma unroll
    for (int r = 0; r < 8; ++r) {
        float z = acc[r] + bv;
        if (relu) z = fmaxf(z, 0.0f);
        out[(size_t)(m0 + r + 8 * hi) * DIM + nc] = z;
    }
}

// ------------------------------------------------- final layer (dout = 3) ----
// Wave per node: lane L accumulates k = L + 32j (j=0..7), shuffle-reduce.
__global__ void out_kernel(const float* __restrict__ agg, const float* __restrict__ hin,
                           const float* __restrict__ W3,  const float* __restrict__ b3,
                           float* __restrict__ out) {
    int tid  = blockIdx.x * blockDim.x + threadIdx.x;
    int node = tid >> 5;
    int lane = tid & 31;
    if (node >= N_NODES) return;

    float s0 = 0.f, s1 = 0.f, s2 = 0.f;
    #pragma unroll
    for (int j = 0; j < 8; ++j) {
        int   k = lane + 32 * j;
        float x = (k < DIM) ? agg[(size_t)node * DIM + k]
                            : hin[(size_t)node * DIM + (k - DIM)];
        s0 += x * W3[k * OUTD + 0];
        s1 += x * W3[k * OUTD + 1];
        s2 += x * W3[k * OUTD + 2];
    }
    #pragma unroll
    for (int off = 16; off > 0; off >>= 1) {
        s0 += __shfl_down(s0, off, 32);
        s1 += __shfl_down(s1, off, 32);
        s2 += __shfl_down(s2, off, 32);
    }
    if (lane == 0) {
        out[(size_t)node * OUTD + 0] = s0 + b3[0];
        out[(size_t)node * OUTD + 1] = s1 + b3[1];
        out[(size_t)node * OUTD + 2] = s2 + b3[2];
    }
}

// ---------------------------------------------------------------------------
extern "C" void kernel_launch(void* const* d_in, const int* in_sizes, int n_in,
                              void* d_out, int out_size, void* d_ws, size_t ws_size,
                              hipStream_t stream) {
    (void)in_sizes; (void)n_in; (void)out_size; (void)ws_size;

    const float* h  = (const float*)d_in[0];
    const int*   ei = (const int*)d_in[1];          // [2, E] int32 (harness dtype rule)
    const float* W0 = (const float*)d_in[2];
    const float* b0 = (const float*)d_in[3];
    const float* W1 = (const float*)d_in[4];
    const float* b1 = (const float*)d_in[5];
    const float* W2 = (const float*)d_in[6];
    const float* b2 = (const float*)d_in[7];
    const float* W3 = (const float*)d_in[8];
    const float* b3 = (const float*)d_in[9];

    const int* row = ei;             // edge_index[0]
    const int* col = ei + N_EDGES;   // edge_index[1]

    // Workspace carve-out (floats): norm[E] | dis[N] | agg[N*128] | hA | hB
    float* ws   = (float*)d_ws;
    float* norm = ws;
    float* dis  = norm + N_EDGES;
    float* agg  = dis + N_NODES;
    float* hA   = agg + (size_t)N_NODES * DIM;
    float* hB   = hA  + (size_t)N_NODES * DIM;
    float* outp = (float*)d_out;

    // Degree -> 1/sqrt(deg) -> per-edge symmetric norm.
    hipMemsetAsync(dis, 0, N_NODES * sizeof(float), stream);
    deg_kernel <<<(N_EDGES + 255) / 256, 256, 0, stream>>>(col, dis);
    dis_kernel <<<(N_NODES + 255) / 256, 256, 0, stream>>>(dis);
    norm_kernel<<<(N_EDGES + 255) / 256, 256, 0, stream>>>(row, col, dis, norm);

    const float* lin[4]  = { h,  hA, hB, hA };
    float*       lout[3] = { hA, hB, hA };
    const float* Ws[4]   = { W0, W1, W2, W3 };
    const float* bs[4]   = { b0, b1, b2, b3 };

    for (int l = 0; l < 4; ++l) {
        hipMemsetAsync(agg, 0, (size_t)N_NODES * DIM * sizeof(float), stream);
        scatter_kernel<<<(N_EDGES * 32) / 256, 256, 0, stream>>>(lin[l], norm, row, col, agg);
        if (l < 3) {
            gemm_kernel<<<N_NODES / 16, 256, 0, stream>>>(agg, lin[l], Ws[l], bs[l], lout[l], 1);
        } else {
            out_kernel<<<(N_NODES * 32 + 255) / 256, 256, 0, stream>>>(agg, lin[l], Ws[l], bs[l], outp);
        }
    }
}